// TAGConv_56908316672631
// MI455X (gfx1250) — compile-verified
//
#include <hip/hip_runtime.h>
#include <hip/hip_bf16.h>

typedef __attribute__((ext_vector_type(2))) float v2f;
typedef __attribute__((ext_vector_type(8))) float v8f;

#define D 64            // feature dim (D_IN == D_OUT == 64)
#define KTOT 256        // (K+1)*D_IN

// ---------------- degree: atomic count of incoming edges at col ----------------
__global__ void deg_kernel(const long long* __restrict__ ei, float* __restrict__ deg,
                           int E_) {
    int e = blockIdx.x * blockDim.x + threadIdx.x;
    if (e >= E_) return;
    long long c = ei[(long long)E_ + e];
    atomicAdd(&deg[c], 1.0f);
}

// ---------------- dinv = deg > 0 ? rsqrt(deg) : 0 (in place) ----------------
__global__ void dinv_kernel(float* __restrict__ deg, int Nn) {
    int i = blockIdx.x * blockDim.x + threadIdx.x;
    if (i >= Nn) return;
    float d = deg[i];
    deg[i] = (d > 0.0f) ? rsqrtf(d) : 0.0f;
}

// ---------------- norm[e] = dinv[row]*dinv[col] ----------------
__global__ void norm_kernel(const long long* __restrict__ ei,
                            const float* __restrict__ dinv,
                            float* __restrict__ norm, int E_) {
    int e = blockIdx.x * blockDim.x + threadIdx.x;
    if (e >= E_) return;
    long long r = ei[e];
    long long c = ei[(long long)E_ + e];
    norm[e] = dinv[r] * dinv[c];
}

// ---------------- one propagation hop: dst[col] += src[row]*norm ----------------
// 16 lanes per edge, each lane handles 4 consecutive features (float4 gather).
__global__ void prop_kernel(const float* __restrict__ src, float* __restrict__ dst,
                            const long long* __restrict__ ei,
                            const float* __restrict__ norm, int E_) {
    long long t = (long long)blockIdx.x * blockDim.x + threadIdx.x;
    int e  = (int)(t >> 4);
    if (e >= E_) return;
    int fq = (int)(t & 15) * 4;
    long long r = ei[e];
    long long c = ei[(long long)E_ + e];
    float nrm = norm[e];
    const float4 v = *(const float4*)(src + r * D + fq);
    float* d = dst + c * D + fq;
    atomicAdd(d + 0, v.x * nrm);
    atomicAdd(d + 1, v.y * nrm);
    atomicAdd(d + 2, v.z * nrm);
    atomicAdd(d + 3, v.w * nrm);
}

// ---------------- GEMM tail: out = [x|h1|h2|h3] @ W + b via f32 WMMA ----------------
// 256 threads = 8 waves; each wave computes a 16(M) x 64(N) tile of the output
// using V_WMMA_F32_16X16X4_F32. W is staged in LDS pre-paired along K so each
// B-fragment is one aligned ds_load_b64 into an even VGPR pair.
__global__ void __launch_bounds__(256)
gemm_kernel(const float* __restrict__ x,  const float* __restrict__ h1,
            const float* __restrict__ h2, const float* __restrict__ h3,
            const float* __restrict__ W,  const float* __restrict__ bias,
            float* __restrict__ out, int Nn) {
    // ldsB[kp*D + col] = { W[2kp][col], W[2kp+1][col] }   (kp in [0,128))
    __shared__ v2f ldsB[(KTOT / 2) * D];   // 65536 bytes

    for (int j = threadIdx.x; j < (KTOT / 2) * D; j += blockDim.x) {
        const int kp  = j >> 6;         // k-pair index
        const int col = j & (D - 1);
        v2f b;
        b.x = W[(2 * kp)     * D + col];
        b.y = W[(2 * kp + 1) * D + col];
        ldsB[j] = b;
    }
    __syncthreads();

    const int lane = threadIdx.x & 31;
    const int wave = threadIdx.x >> 5;
    const int ml   = lane & 15;
    const int hi   = (lane < 16) ? 0 : 1;     // lane half
    const int koff = hi * 2;                  // A/B frag K offset per ISA layout
    const int m0   = (blockIdx.x * 8 + wave) * 16;

    // clamped source row for A-fragment loads (garbage rows never stored)
    long long mrow = m0 + ml;
    if (mrow >= Nn) mrow = Nn - 1;

    v8f acc[4] = {v8f{0,0,0,0,0,0,0,0}, v8f{0,0,0,0,0,0,0,0},
                  v8f{0,0,0,0,0,0,0,0}, v8f{0,0,0,0,0,0,0,0}};

    const float* srcs[4] = {x, h1, h2, h3};
    #pragma unroll
    for (int s = 0; s < 4; ++s) {
        const float* sp = srcs[s] + mrow * D;   // constant after full unroll -> global loads
        const int kbase = s * D;
        #pragma unroll
        for (int kb = 0; kb < D; kb += 4) {
            const int kc = kb + koff;
            // A fragment: 16x4 f32; lanes 0-15 hold K=kb+0,1; lanes 16-31 K=kb+2,3
            const v2f a = *(const v2f*)(sp + kc);
            const int kp = (kbase + kc) >> 1;   // even -> exact pair index
            #pragma unroll
            for (int n = 0; n < 4; ++n) {
                const int col = n * 16 + ml;
                const v2f b = ldsB[kp * D + col];   // single ds_load_b64
                acc[n] = __builtin_amdgcn_wmma_f32_16x16x4_f32(
                    false, a, false, b, (short)0, acc[n], false, false);
            }
        }
    }

    // C/D layout: VGPR i -> row (i + 8*hi), lane%16 -> col
    #pragma unroll
    for (int n = 0; n < 4; ++n) {
        const int col = n * 16 + ml;
        const float bv = bias[col];
        #pragma unroll
        for (int i = 0; i < 8; ++i) {
            const int r = m0 + i + 8 * hi;
            if (r < Nn) out[(long long)r * D + col] = acc[n][i] + bv;
        }
    }
}

extern "C" void kernel_launch(void* const* d_in, const int* in_sizes, int n_in,
                              void* d_out, int out_size, void* d_ws, size_t ws_size,
                              hipStream_t stream) {
    const float*     x    = (const float*)d_in[0];
    const long long* ei   = (const long long*)d_in[1];   // int64 per reference
    const float*     W    = (const float*)d_in[2];
    const float*     bias = (const float*)d_in[3];
    float*           out  = (float*)d_out;

    const int Nn = in_sizes[0] / D;     // 100000
    const int E_ = in_sizes[1] / 2;     // 1600000

    // workspace layout (floats): h1 | h2 | h3 | deg/dinv | norm
    float* h1   = (float*)d_ws;
    float* h2   = h1 + (size_t)Nn * D;
    float* h3   = h2 + (size_t)Nn * D;
    float* deg  = h3 + (size_t)Nn * D;
    float* norm = deg + Nn;

    // zero h1..h3 and deg in one contiguous async memset (graph-capture safe)
    hipMemsetAsync(h1, 0, ((size_t)3 * Nn * D + Nn) * sizeof(float), stream);

    const int T = 256;
    deg_kernel <<<(E_ + T - 1) / T, T, 0, stream>>>(ei, deg, E_);
    dinv_kernel<<<(Nn + T - 1) / T, T, 0, stream>>>(deg, Nn);
    norm_kernel<<<(E_ + T - 1) / T, T, 0, stream>>>(ei, deg, norm, E_);

    const long long pt = (long long)E_ * 16;
    const int pgrid = (int)((pt + T - 1) / T);
    prop_kernel<<<pgrid, T, 0, stream>>>(x,  h1, ei, norm, E_);
    prop_kernel<<<pgrid, T, 0, stream>>>(h1, h2, ei, norm, E_);
    prop_kernel<<<pgrid, T, 0, stream>>>(h2, h3, ei, norm, E_);

    // GEMM tail: 8 waves/block * 16 rows/wave = 128 rows per block
    const int gblocks = (Nn + 127) / 128;
    gemm_kernel<<<gblocks, 256, 0, stream>>>(x, h1, h2, h3, W, bias, out, Nn);
}